// GraphAttention_55430847922869
// MI455X (gfx1250) — compile-verified
//
#include <hip/hip_runtime.h>

typedef __attribute__((ext_vector_type(2))) float v2f;
typedef __attribute__((ext_vector_type(8))) float v8f;

#define B_ 8
#define N_ 2048
#define D_ 512
#define P_ 256
#define NSPLIT 32
#define ROWS_PER_SPLIT (N_ / NSPLIT) /* 64 */

/* ws layout (floats):
   [WS_PART, +B_*NSPLIT*D_)  partial row-sums      (131072 floats, 512 KB)
   [WS_XSUM, +16*D_)         padded xsum, M=16     (  8192 floats,  32 KB)
   [WS_COLS, +16*P_)         colsum (GEMM result)  (  4096 floats,  16 KB) */
#define WS_PART 0
#define WS_XSUM (B_ * NSPLIT * D_)
#define WS_COLS (WS_XSUM + 16 * D_)

__global__ void k_partial_reduce(const float* __restrict__ x, float* __restrict__ ws) {
    int d = threadIdx.x;                 // 0..511, coalesced along D
    int b = blockIdx.x / NSPLIT;
    int s = blockIdx.x % NSPLIT;
    const float* xp = x + ((size_t)(b * N_ + s * ROWS_PER_SPLIT)) * D_ + d;
    float acc = 0.0f;
#pragma unroll 8
    for (int n = 0; n < ROWS_PER_SPLIT; ++n)
        acc += xp[(size_t)n * D_];
    ws[WS_PART + (b * NSPLIT + s) * D_ + d] = acc;
}

__global__ void k_final_reduce(float* __restrict__ ws) {
    int d = threadIdx.x;                 // 0..511
    int r = blockIdx.x;                  // 0..15 (rows 8..15 are zero padding)
    float acc = 0.0f;
    if (r < B_) {
        for (int s = 0; s < NSPLIT; ++s)
            acc += ws[WS_PART + (r * NSPLIT + s) * D_ + d];
    }
    ws[WS_XSUM + r * D_ + d] = acc;
}

/* colsum[16,256] = xsum[16,512] @ kernel[512,256] via V_WMMA_F32_16X16X4_F32.
   One wave (32 lanes) per 16-wide N tile; K looped in steps of 4 -> 128 WMMAs. */
__global__ void k_wmma_gemm(const float* __restrict__ kern, float* __restrict__ ws) {
    int lane  = threadIdx.x;             // 0..31, EXEC all ones (no divergence)
    int half  = lane >> 4;               // 0: lanes 0-15, 1: lanes 16-31
    int l16   = lane & 15;
    int ntile = blockIdx.x;              // 0..15
    const float* xsum = ws + WS_XSUM;

    v8f c = {0.f, 0.f, 0.f, 0.f, 0.f, 0.f, 0.f, 0.f};
    for (int k = 0; k < D_; k += 4) {
        int ka = k + 2 * half;           // A: VGPR0 holds K=0/K=2 halves, VGPR1 K=1/K=3
        v2f a, bm;
        a.x  = xsum[l16 * D_ + ka];      // A 16x4: lane%16 = M
        a.y  = xsum[l16 * D_ + ka + 1];
        bm.x = kern[(size_t)ka * P_ + ntile * 16 + l16];        // B 4x16: lane%16 = N
        bm.y = kern[(size_t)(ka + 1) * P_ + ntile * 16 + l16];
        c = __builtin_amdgcn_wmma_f32_16x16x4_f32(
                /*neg_a=*/false, a, /*neg_b=*/false, bm,
                /*c_mod=*/(short)0, c, /*reuse_a=*/false, /*reuse_b=*/false);
    }
#pragma unroll
    for (int r = 0; r < 8; ++r) {        // C/D: VGPR r -> row r (+8 for upper lane half)
        int m = r + 8 * half;
        ws[WS_COLS + m * P_ + ntile * 16 + l16] = c[r];
    }
}

__global__ void k_broadcast(const float* __restrict__ ws, float4* __restrict__ out) {
    const float4* cols = (const float4*)(ws + WS_COLS);
    size_t idx = (size_t)blockIdx.x * blockDim.x + threadIdx.x; // 0..1048575
    int    p4  = (int)(idx & 63);        // P_/4 = 64 float4 per row
    size_t bi  = idx >> 6;               // b*N + i
    int    b   = (int)(bi / N_);
    out[idx] = cols[b * (P_ / 4) + p4];  // colsum stays hot in L2
}

extern "C" void kernel_launch(void* const* d_in, const int* in_sizes, int n_in,
                              void* d_out, int out_size, void* d_ws, size_t ws_size,
                              hipStream_t stream) {
    (void)in_sizes; (void)n_in; (void)out_size; (void)ws_size;
    const float* x    = (const float*)d_in[0];  // (B,N,D) fp32
    const float* kern = (const float*)d_in[1];  // (D,P)   fp32
    /* d_in[2] (wa) and d_in[3] (ba) are provably dead: softmax over a
       singleton axis makes alpha == 1 exactly. */
    float* ws  = (float*)d_ws;
    float4* out4 = (float4*)d_out;

    k_partial_reduce<<<B_ * NSPLIT, D_, 0, stream>>>(x, ws);
    k_final_reduce  <<<16, D_, 0, stream>>>(ws);
    k_wmma_gemm     <<<P_ / 16, 32, 0, stream>>>(kern, ws);
    k_broadcast     <<<(B_ * N_ * P_ / 4) / 256, 256, 0, stream>>>(ws, out4);
}